// SPVCNN_80178449482543
// MI455X (gfx1250) — compile-verified
//
#include <hip/hip_runtime.h>
#include <hip/hip_bf16.h>
#include <math.h>

// ---------------- problem constants (from reference) ----------------
#define NP_  200000
#define NV   60000
#define NB   4
#define KK   27
#define C1   4
#define C2   32
#define NCLS 20

// ---------------- workspace layout (float offsets) ----------------
#define OFF_VOX   0u            // NV*4
#define OFF_CNT   240000u       // NV
#define OFF_BSUM  300000u       // NB*32
#define OFF_BMAX  300128u       // NB*32
#define OFF_BCNT  300256u       // NB
#define ZERO_N    300260u       // floats to zero each call
#define OFF_CA    300272u       // NB*32
#define OFF_H1    300400u       // NV*32
#define OFF_H2    2220400u      // NV*32
#define OFF_S2    4140400u      // NV*2
#define OFF_SA    4260400u      // NV

typedef __attribute__((ext_vector_type(2))) float v2f;
typedef __attribute__((ext_vector_type(8))) float v8f;

// D = A(16x4) * B(4x16) + C, fp32 WMMA
__device__ __forceinline__ v8f wmma_f32(v2f a, v2f b, v8f c) {
  return __builtin_amdgcn_wmma_f32_16x16x4_f32(false, a, false, b, (short)0, c, false, false);
}

__device__ __forceinline__ float sigmoidf_(float x) { return 1.0f / (1.0f + expf(-x)); }

// ---------------- 0: zero accumulators ----------------
__global__ void spv_zero(float* ws) {
  unsigned i = blockIdx.x * blockDim.x + threadIdx.x;
  if (i < ZERO_N) ws[i] = 0.0f;
}

// ---------------- 1: voxelize scatter ----------------
__global__ void spv_vox_scatter(const float* __restrict__ pf, const int* __restrict__ p2v,
                                float* __restrict__ vox, float* __restrict__ cnt) {
  int p = blockIdx.x * blockDim.x + threadIdx.x;
  if (p >= NP_) return;
  int v = p2v[p];
  const float4 f = *reinterpret_cast<const float4*>(pf + p * 4);
  atomicAdd(&vox[v * 4 + 0], f.x);
  atomicAdd(&vox[v * 4 + 1], f.y);
  atomicAdd(&vox[v * 4 + 2], f.z);
  atomicAdd(&vox[v * 4 + 3], f.w);
  atomicAdd(&cnt[v], 1.0f);
}

// ---------------- 2: voxelize finalize (divide by count) ----------------
__global__ void spv_vox_final(float* __restrict__ vox, const float* __restrict__ cnt) {
  int v = blockIdx.x * blockDim.x + threadIdx.x;
  if (v >= NV) return;
  float inv = 1.0f / fmaxf(cnt[v], 1.0f);
  vox[v * 4 + 0] *= inv;
  vox[v * 4 + 1] *= inv;
  vox[v * 4 + 2] *= inv;
  vox[v * 4 + 3] *= inv;
}

// ---------------- 3: conv1 (Cin=4 -> Cout=32) + BN + ReLU, WMMA f32 ----------------
__global__ void spv_conv1(const float* __restrict__ vox, const float* __restrict__ W1,
                          const float* __restrict__ g1, const float* __restrict__ b1,
                          const int* __restrict__ nidx, const int* __restrict__ nmask,
                          float* __restrict__ h1) {
  const int lane = threadIdx.x & 31;
  const int wave = threadIdx.x >> 5;
  const int tile = (blockIdx.x * (blockDim.x >> 5) + wave) * 16;
  if (tile >= NV) return;                 // wave-uniform: EXEC stays all-ones
  const int m  = lane & 15;               // row within tile / B,C,D column
  const int hi = lane >> 4;               // 0: K=0,1   1: K=2,3
  const int cb = hi * 2;
  const int v  = tile + m;

  v8f acc0 = {}; v8f acc1 = {};
  for (int k = 0; k < KK; ++k) {
    int   idx = nidx[k * NV + v];
    float msk = nmask[k * NV + v] ? 1.0f : 0.0f;
    v2f a;
    a.x = msk * vox[idx * 4 + cb + 0];
    a.y = msk * vox[idx * 4 + cb + 1];
    const float* wk = W1 + k * C1 * C2;   // [4][32]
    v2f b0, b1v;
    b0.x  = wk[(cb + 0) * C2 + m];
    b0.y  = wk[(cb + 1) * C2 + m];
    b1v.x = wk[(cb + 0) * C2 + m + 16];
    b1v.y = wk[(cb + 1) * C2 + m + 16];
    acc0 = wmma_f32(a, b0,  acc0);
    acc1 = wmma_f32(a, b1v, acc1);
  }
  const float ga = g1[m], ba = b1[m], gb = g1[m + 16], bb = b1[m + 16];
#pragma unroll
  for (int r = 0; r < 8; ++r) {
    int row = tile + r + hi * 8;
    h1[row * C2 + m]      = fmaxf(fmaf(acc0[r], ga, ba), 0.0f);
    h1[row * C2 + m + 16] = fmaxf(fmaf(acc1[r], gb, bb), 0.0f);
  }
}

// ---------------- 4: conv2 (32 -> 32) + BN + ReLU, WMMA f32 ----------------
// W2 (27*32*32 f32 = 108 KB) is staged in LDS once per block via the CDNA5
// async global->LDS copy path (ASYNCcnt), then all 8 waves read B-operands
// from LDS instead of refetching from global memory.
__global__ void spv_conv2(const float* __restrict__ h1, const float* __restrict__ W2,
                          const float* __restrict__ g2, const float* __restrict__ b2,
                          const int* __restrict__ nidx, const int* __restrict__ nmask,
                          float* __restrict__ h2) {
  __shared__ float wlds[KK * C2 * C2];    // 27648 floats = 110592 B (< 320 KB LDS)

  // --- cooperative async staging: 6912 x b128 chunks, 27 per thread ---
  {
    const unsigned t = threadIdx.x;
#pragma unroll
    for (int i = 0; i < KK; ++i) {
      unsigned elem = (t + (unsigned)i * 256u) * 4u;     // float index of 16B chunk
      unsigned ldsAddr = (unsigned)(uintptr_t)(wlds + elem); // LDS aperture: addr[31:0] = LDS offset
      unsigned gByteOff = elem * 4u;                     // byte offset from W2 base
      asm volatile("global_load_async_to_lds_b128 %0, %1, %2"
                   :: "v"(ldsAddr), "v"(gByteOff), "s"(W2) : "memory");
    }
    asm volatile("s_wait_asynccnt 0x0" ::: "memory");
    __syncthreads();
  }

  const int lane = threadIdx.x & 31;
  const int wave = threadIdx.x >> 5;
  const int tile = (blockIdx.x * (blockDim.x >> 5) + wave) * 16;
  if (tile >= NV) return;                 // wave-uniform: EXEC stays all-ones
  const int m  = lane & 15;
  const int hi = lane >> 4;
  const int cb = hi * 2;
  const int v  = tile + m;

  v8f acc0 = {}; v8f acc1 = {};
  for (int k = 0; k < KK; ++k) {
    int   idx = nidx[k * NV + v];
    float msk = nmask[k * NV + v] ? 1.0f : 0.0f;
    float4 row[8];
    const float4* src = reinterpret_cast<const float4*>(h1 + idx * C2);
#pragma unroll
    for (int q = 0; q < 8; ++q) row[q] = src[q];
    const float* wk = wlds + k * C2 * C2; // [32][32] in LDS
#pragma unroll
    for (int ks = 0; ks < 8; ++ks) {      // K-slices of 4 channels
      v2f a;
      a.x = msk * (hi ? row[ks].z : row[ks].x);
      a.y = msk * (hi ? row[ks].w : row[ks].y);
      v2f b0, b1v;
      b0.x  = wk[(ks * 4 + cb + 0) * C2 + m];
      b0.y  = wk[(ks * 4 + cb + 1) * C2 + m];
      b1v.x = wk[(ks * 4 + cb + 0) * C2 + m + 16];
      b1v.y = wk[(ks * 4 + cb + 1) * C2 + m + 16];
      acc0 = wmma_f32(a, b0,  acc0);
      acc1 = wmma_f32(a, b1v, acc1);
    }
  }
  const float ga = g2[m], ba = b2[m], gb = g2[m + 16], bb = b2[m + 16];
#pragma unroll
  for (int r = 0; r < 8; ++r) {
    int row = tile + r + hi * 8;
    h2[row * C2 + m]      = fmaxf(fmaf(acc0[r], ga, ba), 0.0f);
    h2[row * C2 + m + 16] = fmaxf(fmaf(acc1[r], gb, bb), 0.0f);
  }
}

// ---------------- 5: per-batch sum / max pooling (atomics) ----------------
__global__ void spv_pool(const float* __restrict__ h2, const int* __restrict__ vb,
                         float* __restrict__ bsum, float* __restrict__ bmax,
                         float* __restrict__ bcnt) {
  unsigned i = blockIdx.x * blockDim.x + threadIdx.x;
  if (i >= (unsigned)(NV * C2)) return;
  int v = i >> 5, c = i & 31;
  int b = vb[v];
  float x = h2[i];
  atomicAdd(&bsum[b * C2 + c], x);
  // post-ReLU values are >= 0, so uint bit-pattern max == float max (init 0)
  atomicMax(reinterpret_cast<unsigned int*>(bmax) + b * C2 + c, __float_as_uint(x));
  if (c == 0) atomicAdd(&bcnt[b], 1.0f);
}

// ---------------- 6: channel-attention MLP (tiny, single block) ----------------
__global__ void spv_ca_mlp(const float* __restrict__ bsum, const float* __restrict__ bmax,
                           const float* __restrict__ bcnt,
                           const float* __restrict__ w1, const float* __restrict__ bb1,
                           const float* __restrict__ w2, const float* __restrict__ bb2,
                           float* __restrict__ ca) {
  __shared__ float hidA[NB][8], hidM[NB][8];
  int t = threadIdx.x;
  if (t < NB * 8) {
    int b = t >> 3, j = t & 7;
    float invc = 1.0f / fmaxf(bcnt[b], 1.0f);
    float sA = bb1[j], sM = bb1[j];
    for (int c = 0; c < C2; ++c) {
      float w = w1[c * 8 + j];
      sA = fmaf(bsum[b * C2 + c] * invc, w, sA);
      sM = fmaf(bmax[b * C2 + c],        w, sM);
    }
    hidA[b][j] = fmaxf(sA, 0.0f);
    hidM[b][j] = fmaxf(sM, 0.0f);
  }
  __syncthreads();
  if (t < NB * C2) {
    int b = t >> 5, c = t & 31;
    float s = 2.0f * bb2[c];              // mlp(avg)+mlp(max) each adds bb2
    for (int j = 0; j < 8; ++j)
      s = fmaf(hidA[b][j] + hidM[b][j], w2[j * C2 + c], s);
    ca[b * C2 + c] = sigmoidf_(s);
  }
}

// ---------------- 7: apply channel attention + build s2 = [mean,max] ----------------
__global__ void spv_apply_ca(float* __restrict__ h2, const int* __restrict__ vb,
                             const float* __restrict__ ca, float* __restrict__ s2) {
  int v = blockIdx.x * blockDim.x + threadIdx.x;
  if (v >= NV) return;
  const float4* cav = reinterpret_cast<const float4*>(ca + vb[v] * C2);
  float4* hv = reinterpret_cast<float4*>(h2 + v * C2);
  float sum = 0.0f, mx = -3.4e38f;
#pragma unroll
  for (int q = 0; q < 8; ++q) {
    float4 hh = hv[q], cc = cav[q];
    hh.x *= cc.x; hh.y *= cc.y; hh.z *= cc.z; hh.w *= cc.w;
    hv[q] = hh;
    sum += hh.x + hh.y + hh.z + hh.w;
    mx = fmaxf(mx, fmaxf(fmaxf(hh.x, hh.y), fmaxf(hh.z, hh.w)));
  }
  s2[v * 2 + 0] = sum * (1.0f / (float)C2);
  s2[v * 2 + 1] = mx;
}

// ---------------- 8: spatial attention sparse conv (2 -> 1) + sigmoid ----------------
__global__ void spv_sa(const float* __restrict__ s2, const float* __restrict__ saW,
                       const int* __restrict__ nidx, const int* __restrict__ nmask,
                       float* __restrict__ sa) {
  int v = blockIdx.x * blockDim.x + threadIdx.x;
  if (v >= NV) return;
  float acc = 0.0f;
  for (int k = 0; k < KK; ++k) {
    int   idx = nidx[k * NV + v];
    float msk = nmask[k * NV + v] ? 1.0f : 0.0f;
    acc = fmaf(msk * s2[idx * 2 + 0], saW[k * 2 + 0], acc);
    acc = fmaf(msk * s2[idx * 2 + 1], saW[k * 2 + 1], acc);
  }
  sa[v] = sigmoidf_(acc);
}

// ---------------- 9: gather + classifier (32 -> 20), WMMA f32 ----------------
__global__ void spv_cls(const float* __restrict__ h2, const float* __restrict__ sa,
                        const int* __restrict__ p2v, const float* __restrict__ cw,
                        const float* __restrict__ cbias, float* __restrict__ out) {
  const int lane = threadIdx.x & 31;
  const int wave = threadIdx.x >> 5;
  const int tile = (blockIdx.x * (blockDim.x >> 5) + wave) * 16;
  if (tile >= NP_) return;
  const int m  = lane & 15;
  const int hi = lane >> 4;
  const int cb = hi * 2;
  const int p  = tile + m;

  int   v  = p2v[p];
  float sc = sa[v];                       // fold spatial attention into gathered row
  float4 row[8];
  const float4* src = reinterpret_cast<const float4*>(h2 + v * C2);
#pragma unroll
  for (int q = 0; q < 8; ++q) row[q] = src[q];

  v8f acc0 = {}; v8f acc1 = {};
#pragma unroll
  for (int ks = 0; ks < 8; ++ks) {
    v2f a;
    a.x = sc * (hi ? row[ks].z : row[ks].x);
    a.y = sc * (hi ? row[ks].w : row[ks].y);
    v2f b0, b1v;
    b0.x  = cw[(ks * 4 + cb + 0) * NCLS + m];
    b0.y  = cw[(ks * 4 + cb + 1) * NCLS + m];
    bool ok = (m + 16) < NCLS;
    b1v.x = ok ? cw[(ks * 4 + cb + 0) * NCLS + m + 16] : 0.0f;
    b1v.y = ok ? cw[(ks * 4 + cb + 1) * NCLS + m + 16] : 0.0f;
    acc0 = wmma_f32(a, b0,  acc0);
    acc1 = wmma_f32(a, b1v, acc1);
  }
  const float bia = cbias[m];
  const float bib = (m + 16) < NCLS ? cbias[m + 16] : 0.0f;
#pragma unroll
  for (int r = 0; r < 8; ++r) {
    int pr = tile + r + hi * 8;
    out[pr * NCLS + m] = acc0[r] + bia;
    if (m + 16 < NCLS) out[pr * NCLS + m + 16] = acc1[r] + bib;
  }
}

// ---------------- host-side launcher ----------------
extern "C" void kernel_launch(void* const* d_in, const int* in_sizes, int n_in,
                              void* d_out, int out_size, void* d_ws, size_t ws_size,
                              hipStream_t stream) {
  const float* pf    = (const float*)d_in[0];
  const float* W1    = (const float*)d_in[1];
  const float* g1    = (const float*)d_in[2];
  const float* b1    = (const float*)d_in[3];
  const float* W2    = (const float*)d_in[4];
  const float* g2    = (const float*)d_in[5];
  const float* b2    = (const float*)d_in[6];
  const float* ca_w1 = (const float*)d_in[7];
  const float* ca_b1 = (const float*)d_in[8];
  const float* ca_w2 = (const float*)d_in[9];
  const float* ca_b2 = (const float*)d_in[10];
  const float* saW   = (const float*)d_in[11];
  const float* cw    = (const float*)d_in[12];
  const float* cbias = (const float*)d_in[13];
  const int*   p2v   = (const int*)d_in[14];
  const int*   vb    = (const int*)d_in[15];
  const int*   nidx  = (const int*)d_in[16];
  const int*   nmask = (const int*)d_in[17];
  float*       out   = (float*)d_out;
  float*       ws    = (float*)d_ws;

  float* vox  = ws + OFF_VOX;
  float* cnt  = ws + OFF_CNT;
  float* bsum = ws + OFF_BSUM;
  float* bmax = ws + OFF_BMAX;
  float* bcnt = ws + OFF_BCNT;
  float* ca   = ws + OFF_CA;
  float* h1   = ws + OFF_H1;
  float* h2   = ws + OFF_H2;
  float* s2   = ws + OFF_S2;
  float* sav  = ws + OFF_SA;

  spv_zero<<<(ZERO_N + 255) / 256, 256, 0, stream>>>(ws);
  spv_vox_scatter<<<(NP_ + 255) / 256, 256, 0, stream>>>(pf, p2v, vox, cnt);
  spv_vox_final<<<(NV + 255) / 256, 256, 0, stream>>>(vox, cnt);

  // 8 waves/block, one 16-row WMMA tile per wave
  const int convBlocks = (NV / 16 + 7) / 8;           // 469
  spv_conv1<<<convBlocks, 256, 0, stream>>>(vox, W1, g1, b1, nidx, nmask, h1);
  spv_conv2<<<convBlocks, 256, 0, stream>>>(h1, W2, g2, b2, nidx, nmask, h2);

  spv_pool<<<(NV * C2 + 255) / 256, 256, 0, stream>>>(h2, vb, bsum, bmax, bcnt);
  spv_ca_mlp<<<1, 128, 0, stream>>>(bsum, bmax, bcnt, ca_w1, ca_b1, ca_w2, ca_b2, ca);
  spv_apply_ca<<<(NV + 255) / 256, 256, 0, stream>>>(h2, vb, ca, s2);
  spv_sa<<<(NV + 255) / 256, 256, 0, stream>>>(s2, saW, nidx, nmask, sav);

  const int clsBlocks = (NP_ / 16 + 7) / 8;           // 1563
  spv_cls<<<clsBlocks, 256, 0, stream>>>(h2, sav, p2v, cw, cbias, out);
}